// Self_Attention3D_44032004718621
// MI455X (gfx1250) — compile-verified
//
#include <hip/hip_runtime.h>
#include <hip/hip_bf16.h>

// ---------------------------------------------------------------------------
// Self-attention over 3D volume as register-blocked WMMA GEMMs on gfx1250.
//   B=2, N=D*H*W=4096, C=256, Cq=C/4=64
//   f = x@Wf [B,N,64], g = x@Wg [B,N,64], hT = (x@Wh)^T [B,256,N]
//   s = g@f^T [B,N,N] -> row softmax in place -> beta (tail of d_out)
//   out = gamma * (beta@h) + x                  (head of d_out)
// All matmuls use V_WMMA_F32_16X16X4_F32 (fp32-exact path, matches fp32 ref).
// beta@h additionally stages its shared hT operand in LDS across a 4-wave
// workgroup (bank-padded), cutting its L2 traffic 4x.
// ---------------------------------------------------------------------------

typedef float v2f __attribute__((ext_vector_type(2)));
typedef float v8f __attribute__((ext_vector_type(8)));

#define BATCH 2
#define NPOS  4096      // D*H*W
#define CDIM  256
#define CQ    64        // C/4

__device__ __forceinline__ v8f wmma_f32x4(v2f a, v2f b, v8f c) {
  // (neg_a, A, neg_b, B, c_mod, C, reuse_a, reuse_b)
  return __builtin_amdgcn_wmma_f32_16x16x4_f32(
      false, a, false, b, (short)0, c, false, false);
}

// ---------------------------------------------------------------------------
// Fused f & g projections: one wave per 16-row tile computes the full 64-wide
// f AND g outputs (8 accumulators). The A fragment (x) is loaded once per
// k-step and reused by 8 wmma ops.
// ---------------------------------------------------------------------------
__global__ void proj_fg_kernel(const float* __restrict__ X,
                               const float* __restrict__ Wf,
                               const float* __restrict__ Wg,
                               float* __restrict__ f,
                               float* __restrict__ g) {
  const int mt   = blockIdx.x;      // 0..511 (B*N/16 row tiles)
  const int l    = threadIdx.x;
  const int mn   = l & 15;
  const int half = l >> 4;

  const float* __restrict__ xrow = X + (size_t)(mt * 16 + mn) * CDIM;

  v8f accf[4] = {};
  v8f accg[4] = {};
#pragma unroll 4
  for (int k = 0; k < CDIM; k += 4) {
    const int ko = k + 2 * half;
    v2f a = *(const v2f*)(xrow + ko);
#pragma unroll
    for (int j = 0; j < 4; ++j) {
      const float* wp = Wf + (size_t)ko * CQ + j * 16 + mn;
      v2f bq; bq.x = wp[0]; bq.y = wp[CQ];
      accf[j] = wmma_f32x4(a, bq, accf[j]);
    }
#pragma unroll
    for (int j = 0; j < 4; ++j) {
      const float* wp = Wg + (size_t)ko * CQ + j * 16 + mn;
      v2f bq; bq.x = wp[0]; bq.y = wp[CQ];
      accg[j] = wmma_f32x4(a, bq, accg[j]);
    }
  }
#pragma unroll
  for (int j = 0; j < 4; ++j)
#pragma unroll
    for (int v = 0; v < 8; ++v) {
      size_t idx = (size_t)(mt * 16 + 8 * half + v) * CQ + j * 16 + mn;
      f[idx] = accf[j][v];
      g[idx] = accg[j][v];
    }
}

// ---------------------------------------------------------------------------
// h projection, stored TRANSPOSED: hT[b, c, n] = (x @ Wh)[b, n, c].
// 1x4 column blocking. Transposed store makes the B operand of beta@h a
// contiguous float2 per lane later.
// ---------------------------------------------------------------------------
__global__ void proj_h_kernel(const float* __restrict__ X,
                              const float* __restrict__ Wh,
                              float* __restrict__ hT) {
  const int ngrp = CDIM / 64;                 // 4 column groups of 64
  const int mt   = blockIdx.x / ngrp;         // row tile (0..511)
  const int ng   = blockIdx.x % ngrp;         // column group
  const int l    = threadIdx.x;
  const int mn   = l & 15;
  const int half = l >> 4;

  const float* __restrict__ xrow = X + (size_t)(mt * 16 + mn) * CDIM;

  v8f acc[4] = {};
#pragma unroll 4
  for (int k = 0; k < CDIM; k += 4) {
    const int ko = k + 2 * half;
    v2f a = *(const v2f*)(xrow + ko);
#pragma unroll
    for (int j = 0; j < 4; ++j) {
      const float* wp = Wh + (size_t)ko * CDIM + ng * 64 + j * 16 + mn;
      v2f bq; bq.x = wp[0]; bq.y = wp[CDIM];
      acc[j] = wmma_f32x4(a, bq, acc[j]);
    }
  }

  const int b = (mt * 16) / NPOS;             // batch of this row tile
  const int r0 = mt * 16 - b * NPOS;          // row within batch
#pragma unroll
  for (int j = 0; j < 4; ++j) {
    const int col = ng * 64 + j * 16 + mn;
    float* __restrict__ hcol = hT + ((size_t)b * CDIM + col) * NPOS;
#pragma unroll
    for (int v = 0; v < 8; ++v)
      hcol[r0 + 8 * half + v] = acc[j][v];
  }
}

// ---------------------------------------------------------------------------
// Raw logits s = g @ f^T (K=64), 2x2 register blocking: 4 b64 loads feed
// 4 wmma per k-step. f^T operand layout = contiguous read of row-major f.
// grid = (N/32, N/32, B)
// ---------------------------------------------------------------------------
__global__ void attn_scores_kernel(const float* __restrict__ g,
                                   const float* __restrict__ f,
                                   float* __restrict__ s) {
  const int nt = blockIdx.x;   // row pair (g rows)
  const int mt = blockIdx.y;   // col pair (f rows)
  const int b  = blockIdx.z;

  const int l    = threadIdx.x;
  const int mn   = l & 15;
  const int half = l >> 4;

  const float* __restrict__ g0 = g + ((size_t)b * NPOS + nt * 32 + mn) * CQ;
  const float* __restrict__ g1 = g0 + (size_t)16 * CQ;
  const float* __restrict__ f0 = f + ((size_t)b * NPOS + mt * 32 + mn) * CQ;
  const float* __restrict__ f1 = f0 + (size_t)16 * CQ;

  v8f acc[2][2] = {};
#pragma unroll
  for (int k = 0; k < CQ; k += 4) {
    const int ko = k + 2 * half;
    v2f a0 = *(const v2f*)(g0 + ko);
    v2f a1 = *(const v2f*)(g1 + ko);
    v2f b0 = *(const v2f*)(f0 + ko);
    v2f b1 = *(const v2f*)(f1 + ko);
    acc[0][0] = wmma_f32x4(a0, b0, acc[0][0]);
    acc[0][1] = wmma_f32x4(a0, b1, acc[0][1]);
    acc[1][0] = wmma_f32x4(a1, b0, acc[1][0]);
    acc[1][1] = wmma_f32x4(a1, b1, acc[1][1]);
  }

  float* __restrict__ sb = s + (size_t)b * NPOS * NPOS;
#pragma unroll
  for (int i = 0; i < 2; ++i)
#pragma unroll
    for (int j = 0; j < 2; ++j)
#pragma unroll
      for (int v = 0; v < 8; ++v)
        sb[(size_t)(nt * 32 + i * 16 + 8 * half + v) * NPOS +
           mt * 32 + j * 16 + mn] = acc[i][j][v];
}

// ---------------------------------------------------------------------------
// In-place softmax over each length-N row (beta stays hot in the 192MB L2).
// ---------------------------------------------------------------------------
__global__ void softmax_rows_kernel(float* __restrict__ s) {
  __shared__ float red[256];
  float* __restrict__ p = s + (size_t)blockIdx.x * NPOS;
  const int t = threadIdx.x;

  float m = -__builtin_inff();
  for (int i = t; i < NPOS; i += 256) m = fmaxf(m, p[i]);
  red[t] = m;
  __syncthreads();
  for (int off = 128; off > 0; off >>= 1) {
    if (t < off) red[t] = fmaxf(red[t], red[t + off]);
    __syncthreads();
  }
  m = red[0];
  __syncthreads();

  float sum = 0.f;
  for (int i = t; i < NPOS; i += 256) {
    float e = __expf(p[i] - m);
    p[i] = e;
    sum += e;
  }
  red[t] = sum;
  __syncthreads();
  for (int off = 128; off > 0; off >>= 1) {
    if (t < off) red[t] += red[t + off];
    __syncthreads();
  }
  const float inv = 1.f / red[0];
  for (int i = t; i < NPOS; i += 256) p[i] *= inv;
}

// ---------------------------------------------------------------------------
// out = gamma * (beta @ h) + x, K=4096.
// Block = 4 waves (128 threads); each wave owns 2 row tiles x 4 col tiles.
// The shared 64-col hT chunk (64 k-values, 17KB, stride-68 padding so the 16
// lanes of a ds_load_b64 hit disjoint bank pairs) is staged in LDS once per
// chunk and consumed by all 4 waves: hT L2 traffic drops 4x. beta A-operand
// stays in global (wave-private contiguous b64 rows).
// grid = (N/128, C/64, B).
// ---------------------------------------------------------------------------
#define KC   64          // k-chunk staged per barrier
#define LSTR 68          // LDS row stride in floats (bank padding, 16B aligned)

__global__ void attn_out_kernel(const float* __restrict__ beta,
                                const float* __restrict__ hT,
                                const float* __restrict__ x,
                                const float* __restrict__ gamma,
                                float* __restrict__ out) {
  __shared__ float lb[64 * LSTR];

  const int mtg = blockIdx.x;  // group of 4 row pairs (128 rows)
  const int ng  = blockIdx.y;  // 64-wide column group
  const int b   = blockIdx.z;

  const int tid  = threadIdx.x;     // 0..127
  const int w    = tid >> 5;        // wave id: which row pair
  const int l    = tid & 31;
  const int mn   = l & 15;
  const int half = l >> 4;
  const int mt   = mtg * 4 + w;     // row pair index (0..127)

  const float* __restrict__ b0 =
      beta + (size_t)b * NPOS * NPOS + (size_t)(mt * 32 + mn) * NPOS;
  const float* __restrict__ b1 = b0 + (size_t)16 * NPOS;
  const float* __restrict__ htbase =
      hT + ((size_t)b * CDIM + ng * 64) * NPOS;

  v8f acc[2][4] = {};

  for (int kc = 0; kc < NPOS; kc += KC) {
    __syncthreads();   // protect LDS from previous chunk's readers
    // Cooperative stage: 64 cols x 64 k fp32 (16KB) via float4.
    for (int i = tid; i < 64 * 16; i += 128) {
      const int c = i >> 4;        // column 0..63
      const int q = i & 15;        // float4 index along k
      const float4 vsrc =
          *(const float4*)(htbase + (size_t)c * NPOS + kc + q * 4);
      *(float4*)(&lb[c * LSTR + q * 4]) = vsrc;
    }
    __syncthreads();

#pragma unroll 4
    for (int kk = 0; kk < KC; kk += 4) {
      const int ko = kk + 2 * half;
      v2f a0 = *(const v2f*)(b0 + kc + ko);
      v2f a1 = *(const v2f*)(b1 + kc + ko);
#pragma unroll
      for (int j = 0; j < 4; ++j) {
        v2f bq = *(const v2f*)(&lb[(j * 16 + mn) * LSTR + ko]);
        acc[0][j] = wmma_f32x4(a0, bq, acc[0][j]);
        acc[1][j] = wmma_f32x4(a1, bq, acc[1][j]);
      }
    }
  }

  const float gm = gamma[0];
#pragma unroll
  for (int i = 0; i < 2; ++i)
#pragma unroll
    for (int j = 0; j < 4; ++j)
#pragma unroll
      for (int v = 0; v < 8; ++v) {
        size_t idx = ((size_t)b * NPOS + mt * 32 + i * 16 + 8 * half + v) *
                         (size_t)CDIM + ng * 64 + j * 16 + mn;
        out[idx] = gm * acc[i][j][v] + x[idx];
      }
}

// ---------------------------------------------------------------------------
extern "C" void kernel_launch(void* const* d_in, const int* in_sizes, int n_in,
                              void* d_out, int out_size, void* d_ws, size_t ws_size,
                              hipStream_t stream) {
  const float* x     = (const float*)d_in[0];   // [B, N, C]
  const float* Wf    = (const float*)d_in[1];   // [C, C/4]
  const float* Wg    = (const float*)d_in[2];   // [C, C/4]
  const float* Wh    = (const float*)d_in[3];   // [C, C]
  const float* gamma = (const float*)d_in[4];   // [1]

  float* out  = (float*)d_out;                                  // [B,N,C]
  float* beta = (float*)d_out + (size_t)BATCH * NPOS * CDIM;    // [B,N,N]

  // Workspace: f | g | hT
  float* f  = (float*)d_ws;
  float* g  = f + (size_t)BATCH * NPOS * CQ;
  float* hT = g + (size_t)BATCH * NPOS * CQ;

  const int Mrows = BATCH * NPOS;               // 8192 rows (x as [B*N, C])

  // 1) Projections: fused f&g (x read once), h stored transposed.
  proj_fg_kernel<<<dim3(Mrows / 16), 32, 0, stream>>>(x, Wf, Wg, f, g);
  proj_h_kernel<<<dim3((Mrows / 16) * (CDIM / 64)), 32, 0, stream>>>(x, Wh, hT);

  // 2) Raw logits s = g @ f^T into the beta output region (2x2 blocked).
  attn_scores_kernel<<<dim3(NPOS / 32, NPOS / 32, BATCH), 32, 0, stream>>>(g, f, beta);

  // 3) Row softmax in place -> beta.
  softmax_rows_kernel<<<dim3(BATCH * NPOS), 256, 0, stream>>>(beta);

  // 4) out = gamma * (beta @ h) + x  (4-wave blocks, hT shared via LDS).
  attn_out_kernel<<<dim3(NPOS / 128, CDIM / 64, BATCH), 128, 0, stream>>>(
      beta, hT, x, gamma, out);
}